// VQDiffusionVAE_73581379715513
// MI455X (gfx1250) — compile-verified
//
#include <hip/hip_runtime.h>
#include <cstdint>

typedef __attribute__((ext_vector_type(2))) float v2f;
typedef __attribute__((ext_vector_type(8))) float v8f;

#define K_CB   8192
#define N_PTS  32768
// d_out layout (flat f32): z_q_st | indices | z_q | vq_loss
#define OUT_ZQST 0
#define OUT_IDX  131072
#define OUT_ZQ   163840
#define OUT_LOSS 294912

__global__ void vq_init_kernel(float* out) { out[OUT_LOSS] = 0.0f; }

__global__ __launch_bounds__(256)
void vq_argmin_wmma_kernel(const float* __restrict__ z_e,
                           const float* __restrict__ codebook,
                           float* __restrict__ out) {
  // LDS: [0, 32768) floats = codebook (8192 x 4), [32768, 40960) = ||c||^2
  extern __shared__ float lds[];
  float* cb  = lds;
  float* nrm = lds + K_CB * 4;

  const unsigned t = threadIdx.x;

  // ---- Stage full codebook (128 KB) into LDS with async global->LDS DMA ----
  {
    unsigned ldsBase = (unsigned)(uintptr_t)cb;  // low 32 bits = LDS byte offset
    #pragma unroll
    for (int i = 0; i < 32; ++i) {
      unsigned byteOff = (t + (unsigned)i * 256u) * 16u;  // 16B per thread per iter
      unsigned ldsAddr = ldsBase + byteOff;
      asm volatile("global_load_async_to_lds_b128 %0, %1, %2"
                   :: "v"(ldsAddr), "v"(byteOff), "s"(codebook)
                   : "memory");
    }
    asm volatile("s_wait_asynccnt 0" ::: "memory");
  }
  __syncthreads();

  // ---- Precompute codebook squared norms into LDS ----
  #pragma unroll
  for (int i = 0; i < 32; ++i) {
    int e = (int)t + i * 256;
    const float4 c = *(const float4*)&cb[e * 4];
    nrm[e] = c.x * c.x + c.y * c.y + c.z * c.z + c.w * c.w;
  }
  __syncthreads();

  const int lane  = (int)(t & 31u);
  const int wave  = (int)(t >> 5u);
  const int p     = blockIdx.x * 128 + wave * 16 + (lane & 15); // this lane's point
  const int kBase = (lane < 16) ? 0 : 2;   // which 2 of the 4 components (A/B layout)
  const int rOff  = (lane < 16) ? 0 : 8;   // which 8 rows of C/D this lane holds

  // z_e is [8,4,64,64]; flat point n -> element (b,d): b*16384 + d*4096 + (n&4095)
  const int bImg = p >> 12;
  const int sp   = p & 4095;

  // B operand: 4x16 matrix of the wave's 16 points, pre-scaled by -2
  const float x0 = z_e[bImg * 16384 + kBase * 4096 + sp];
  const float x1 = z_e[bImg * 16384 + (kBase + 1) * 4096 + sp];
  v2f bmat; bmat[0] = -2.0f * x0; bmat[1] = -2.0f * x1;

  float best = 3.4e38f;
  int   bi   = 0;

  // Pointer-bump addressing so the unroller folds addresses into ds offsets
  const float*  aptr = cb + (lane & 15) * 4 + kBase;       // +64 floats / chunk
  const float4* nptr = (const float4*)(nrm + rOff);        // +4 float4s / chunk

  #pragma unroll 8
  for (int e0 = 0; e0 < K_CB; e0 += 16) {
    // A operand: 16 codebook entries x 4 comps (ds_load_b64)
    v2f amat = *(const v2f*)aptr;

    // C operand: ||c||^2 for this lane's 8 rows (2x ds_load_b128, broadcast)
    const float4 n0 = nptr[0];
    const float4 n1 = nptr[1];
    v8f cmat;
    cmat[0] = n0.x; cmat[1] = n0.y; cmat[2] = n0.z; cmat[3] = n0.w;
    cmat[4] = n1.x; cmat[5] = n1.y; cmat[6] = n1.z; cmat[7] = n1.w;

    // D = A*(-2B) + C  ->  score = ||c||^2 - 2 x.c   (argmin-equivalent to d2)
    v8f dmat = __builtin_amdgcn_wmma_f32_16x16x4_f32(
        false, amat, false, bmat, (short)0, cmat, false, false);

    #pragma unroll
    for (int r = 0; r < 8; ++r) {
      float s = dmat[r];
      int   e = e0 + rOff + r;
      if (s < best) { best = s; bi = e; }
    }
    aptr += 64;
    nptr += 4;
  }

  // Combine the two lane-halves (rows 0..7 vs 8..15); prefer lower index on ties
  {
    float ob = __shfl_xor(best, 16, 32);
    int   oi = __shfl_xor(bi, 16, 32);
    if (ob < best || (ob == best && oi < bi)) { best = ob; bi = oi; }
  }

  float lsum = 0.0f;
  if (lane < 16) {
    out[OUT_IDX + p] = (float)bi;
    #pragma unroll
    for (int d = 0; d < 4; ++d) {
      float c = cb[bi * 4 + d];
      float x = z_e[bImg * 16384 + d * 4096 + sp];
      bool  bad = isnan(c) || isinf(c);
      float zq  = bad ? x : c;
      int   o   = bImg * 16384 + d * 4096 + sp;
      out[OUT_ZQST + o] = zq;   // straight-through forward value == z_q
      out[OUT_ZQ + o]   = zq;
      float df = x - zq;
      lsum += df * df;
    }
  }
  // vq_loss = (1 + beta) * mean((x - zq)^2), beta = 0.25, over 131072 elements
  #pragma unroll
  for (int m = 16; m >= 1; m >>= 1) lsum += __shfl_xor(lsum, m, 32);
  if (lane == 0) atomicAdd(&out[OUT_LOSS], lsum * (1.25f / 131072.0f));
}

extern "C" void kernel_launch(void* const* d_in, const int* in_sizes, int n_in,
                              void* d_out, int out_size, void* d_ws, size_t ws_size,
                              hipStream_t stream) {
  const float* z_e      = (const float*)d_in[0];
  const float* codebook = (const float*)d_in[1];
  float* out = (float*)d_out;

  // 160 KB dynamic LDS (codebook + norms); CDNA5 WGP has 320 KB.
  (void)hipFuncSetAttribute((const void*)vq_argmin_wmma_kernel,
                            hipFuncAttributeMaxDynamicSharedMemorySize, 163840);

  hipLaunchKernelGGL(vq_init_kernel, dim3(1), dim3(1), 0, stream, out);
  hipLaunchKernelGGL(vq_argmin_wmma_kernel, dim3(256), dim3(256), 163840, stream,
                     z_e, codebook, out);
}